// GraphStructureLearning_86595130621975
// MI455X (gfx1250) — compile-verified
//
#include <hip/hip_runtime.h>

// ---- problem constants (from reference) ----
#define Bc    16
#define Pc    4096      // P = SEQ + PRED (feature length per node)
#define DMc   64        // N = number of nodes
#define HIDc  4096
#define HSc   4096
#define NHc   16
#define DHc   256       // HS / NH
#define Lc    4         // NLAYER + 1 tokens
#define KTOP  8

typedef __attribute__((ext_vector_type(2))) float v2f;
typedef __attribute__((ext_vector_type(8))) float v8f;

// V_WMMA_F32_16X16X4_F32 : D(16x16 f32) = A(16x4 f32) * B(4x16 f32) + C
__device__ __forceinline__ v8f wmma4(v2f a, v2f b, v8f c) {
  return __builtin_amdgcn_wmma_f32_16x16x4_f32(false, a, false, b, (short)0, c,
                                               false, false);
}

// ---------------- zero-fill (graph-capture safe) ----------------
__global__ void k_zero(float* __restrict__ p, int n) {
  int i = blockIdx.x * 256 + threadIdx.x;
  if (i < n) p[i] = 0.0f;
}

// ---------------- layer-0 Gram: G0 = h h^T, s0 = rowsum(h) ----------------
// h[m,k] = x[b,k,m].  Grid: (pc=32, b=16), 256 threads.
#define PT0 128
__global__ __launch_bounds__(256) void k_gram0(const float* __restrict__ x,
                                               float* __restrict__ G,
                                               float* __restrict__ S) {
  const int pc = blockIdx.x, b = blockIdx.y;
  const float* xb = x + ((size_t)b * Pc + (size_t)pc * PT0) * DMc;
  __shared__ float xs[PT0][DMc];  // [k][m], matches global layout

  const int tid = threadIdx.x;
  // coalesced float4 copy of 128x64 tile
  for (int i = tid; i < PT0 * DMc / 4; i += 256) {
    ((float4*)xs)[i] = ((const float4*)xb)[i];
  }
  __syncthreads();

  // row sums
  {
    int r = tid & 63, part = tid >> 6;  // 4 parts x 32 k each
    float sum = 0.0f;
    for (int k = part * 32; k < part * 32 + 32; k++) sum += xs[k][r];
    atomicAdd(&S[(b * Lc + 0) * DMc + r], sum);
  }

  // Gram via WMMA: 16 tiles (4x4), 8 waves -> 2 tiles each
  const int wave = tid >> 5, lane = tid & 31;
  const int lm = lane & 15, hh = lane >> 4;
  float* Gout = G + (size_t)(b * Lc + 0) * DMc * DMc;
#pragma unroll
  for (int q = 0; q < 2; q++) {
    int t = wave * 2 + q;
    int gm = t >> 2, gn = t & 3;
    v8f gc = {};
    for (int kb = 0; kb < PT0; kb += 4) {
      v2f a, bb;
      a.x = xs[kb + 2 * hh][gm * 16 + lm];
      a.y = xs[kb + 2 * hh + 1][gm * 16 + lm];
      bb.x = xs[kb + 2 * hh][gn * 16 + lm];
      bb.y = xs[kb + 2 * hh + 1][gn * 16 + lm];
      gc = wmma4(a, bb, gc);
    }
#pragma unroll
    for (int r = 0; r < 8; r++)
      atomicAdd(&Gout[(gm * 16 + r + 8 * hh) * DMc + gn * 16 + lm], gc[r]);
  }
}

// ------- fused MLP + Gram: hv = h @ W_l + b_l ; G += hv hv^T ; s += rowsum -------
// Grid: (jc=32, l=3, b=16), 256 threads (8 waves). Jt = 128 output columns per WG.
// LDS: staging xs[64][64] + Wt[64][128] (48KB), overlaid post-loop by hv[64][129].
#define JT 128
__global__ __launch_bounds__(256) void k_mlp_gram(const float* __restrict__ x,
                                                  const float* __restrict__ Wmlp,
                                                  const float* __restrict__ bmlp,
                                                  float* __restrict__ G,
                                                  float* __restrict__ S) {
  const int jc = blockIdx.x, l = blockIdx.y, b = blockIdx.z;
  const int j0 = jc * JT;
  const float* W = Wmlp + (size_t)l * Pc * HIDc;
  const float* xb = x + (size_t)b * Pc * DMc;

  __shared__ float smem[4096 + 64 * JT];              // 48KB
  float (*xs)[DMc] = (float(*)[DMc])smem;             // [k][m], stride 64
  float (*Wt)[JT]  = (float(*)[JT])(smem + 4096);     // [k][j], stride 128
  float (*hv)[JT + 1] = (float(*)[JT + 1])smem;       // overlay (8256 floats)

  const int tid = threadIdx.x;
  const int wave = tid >> 5, lane = tid & 31;
  const int lm = lane & 15, hh = lane >> 4;
  const int mtile = wave >> 1;          // 0..3
  const int ntb = (wave & 1) * 4;       // 4 n-tiles per wave

  v8f acc[4] = {};
  const int m = mtile * 16 + lm;
  for (int kb0 = 0; kb0 < Pc; kb0 += 64) {
    __syncthreads();
    // stage x chunk: 64 rows x 64 floats, contiguous in global
    const float* xsrc = xb + (size_t)kb0 * DMc;
    for (int i = tid; i < 1024; i += 256)
      ((float4*)xs)[i] = ((const float4*)xsrc)[i];
    // stage W chunk: 64 rows x 128 floats (coalesced float4)
    for (int i = tid; i < 2048; i += 256) {
      int r = i >> 5, c4 = i & 31;
      ((float4*)&Wt[r][0])[c4] =
          *(const float4*)&W[(size_t)(kb0 + r) * HIDc + j0 + c4 * 4];
    }
    // prefetch next W chunk into cache (global_prefetch_b8)
    if (kb0 + 64 < Pc) {
      int rr = tid >> 2, cc = (tid & 3) * 32;
      __builtin_prefetch(&W[(size_t)(kb0 + 64 + rr) * HIDc + j0 + cc], 0, 0);
    }
    __syncthreads();
    for (int kk = 0; kk < 64; kk += 4) {
      v2f a;  // one ds_load_2addr_stride64 (row stride = 64 dwords)
      a.x = xs[kk + 2 * hh][m];
      a.y = xs[kk + 2 * hh + 1][m];
#pragma unroll
      for (int t = 0; t < 4; t++) {
        int j = (ntb + t) * 16 + lm;
        v2f bb;  // one ds_load_2addr_stride64 (row stride = 128 dwords)
        bb.x = Wt[kk + 2 * hh][j];
        bb.y = Wt[kk + 2 * hh + 1][j];
        acc[t] = wmma4(a, bb, acc[t]);
      }
    }
  }
  __syncthreads();  // staging buffers dead; overlay hv

  // bias + park hv tile in LDS
#pragma unroll
  for (int t = 0; t < 4; t++) {
    int jl = (ntb + t) * 16 + lm;
    float bv = bmlp[l * HIDc + j0 + jl];
#pragma unroll
    for (int r = 0; r < 8; r++)
      hv[mtile * 16 + r + 8 * hh][jl] = acc[t][r] + bv;
  }
  __syncthreads();

  // partial row sums of hv chunk
  {
    int r = tid & 63, part = tid >> 6;
    float sum = 0.0f;
    for (int jj = part * 32; jj < part * 32 + 32; jj++) sum += hv[r][jj];
    atomicAdd(&S[(b * Lc + (l + 1)) * DMc + r], sum);
  }

  // Gram of this chunk: 16 tiles, 2 per wave
  float* Gout = G + (size_t)(b * Lc + (l + 1)) * DMc * DMc;
#pragma unroll
  for (int q = 0; q < 2; q++) {
    int t = wave * 2 + q;
    int gm = t >> 2, gn = t & 3;
    v8f gc = {};
    for (int kb = 0; kb < JT; kb += 4) {
      v2f a, bb;
      a.x = hv[gm * 16 + lm][kb + 2 * hh];
      a.y = hv[gm * 16 + lm][kb + 2 * hh + 1];
      bb.x = hv[gn * 16 + lm][kb + 2 * hh];
      bb.y = hv[gn * 16 + lm][kb + 2 * hh + 1];
      gc = wmma4(a, bb, gc);
    }
#pragma unroll
    for (int r = 0; r < 8; r++)
      atomicAdd(&Gout[(gm * 16 + r + 8 * hh) * DMc + gn * 16 + lm], gc[r]);
  }
}

// ------- adjacency from Gram stats: top-K mask, relu, symmetrize -> tok -------
// Grid: 64 (= b*4+layer), 64 threads (one per row).
__global__ __launch_bounds__(64) void k_adj(const float* __restrict__ G,
                                            const float* __restrict__ S,
                                            float* __restrict__ tok) {
  const int bl = blockIdx.x;
  const float* g = G + (size_t)bl * DMc * DMc;
  const float* s = S + (size_t)bl * DMc;
  __shared__ float Fm[DMc][DMc + 1];
  __shared__ float norm_s[DMc], mass_s[DMc];

  const int n = threadIdx.x;
  float nn = sqrtf(g[n * DMc + n]);
  norm_s[n] = nn;
  mass_s[n] = s[n] / nn;
  __syncthreads();

  float row[DMc];
  const float mn = mass_s[n];
  const float invn = 1.0f / norm_s[n];
#pragma unroll
  for (int j = 0; j < DMc; j++) {
    float sim = g[n * DMc + j] * invn / norm_s[j];
    row[j] = mn * mass_s[j] * sim * sim;  // G_CONST = 1
  }
  // 8th-largest via 8 strict-max passes
  float prev = __builtin_inff(), kth = -__builtin_inff();
  for (int it = 0; it < KTOP; it++) {
    float mx = -__builtin_inff();
#pragma unroll
    for (int j = 0; j < DMc; j++)
      if (row[j] < prev && row[j] > mx) mx = row[j];
    kth = mx;
    prev = mx;
  }
#pragma unroll
  for (int j = 0; j < DMc; j++) {
    float v = (row[j] >= kth) ? row[j] : 0.0f;
    Fm[n][j] = v > 0.0f ? v : 0.0f;
  }
  __syncthreads();
  float* out = tok + (size_t)bl * DMc * DMc;
  for (int j = 0; j < DMc; j++)
    out[n * DMc + j] = 0.5f * (Fm[n][j] + Fm[j][n]);
}

// ------- generic 64x4096 GEMM: C = A(64x4096) @ W(4096x4096) + bias -------
// LDS: as[64][65] (padded; lanes index rows) + Wt[64][128] staging.
__device__ __forceinline__ void gemm64_body(const float* __restrict__ A,
                                            const float* __restrict__ W,
                                            const float* __restrict__ bias,
                                            float* __restrict__ C, int j0,
                                            bool thresh, float (*as)[65],
                                            float (*Wt)[JT]) {
  const int tid = threadIdx.x;
  const int wave = tid >> 5, lane = tid & 31;
  const int lm = lane & 15, hh = lane >> 4;
  const int mtile = wave >> 1;
  const int ntb = (wave & 1) * 4;

  v8f acc[4] = {};
  for (int kb0 = 0; kb0 < HSc; kb0 += 64) {
    __syncthreads();
    for (int i = tid; i < 64 * 16; i += 256) {  // stage A tile [64m][64k]
      int mrow = i >> 4, c4 = i & 15;
      float4 v = *(const float4*)&A[(size_t)mrow * HSc + kb0 + c4 * 4];
      as[mrow][c4 * 4 + 0] = v.x;
      as[mrow][c4 * 4 + 1] = v.y;
      as[mrow][c4 * 4 + 2] = v.z;
      as[mrow][c4 * 4 + 3] = v.w;
    }
    for (int i = tid; i < 2048; i += 256) {  // stage W tile [64k][128j]
      int r = i >> 5, c4 = i & 31;
      ((float4*)&Wt[r][0])[c4] =
          *(const float4*)&W[(size_t)(kb0 + r) * HSc + j0 + c4 * 4];
    }
    if (kb0 + 64 < HSc) {  // prefetch next W chunk
      int rr = tid >> 2, cc = (tid & 3) * 32;
      __builtin_prefetch(&W[(size_t)(kb0 + 64 + rr) * HSc + j0 + cc], 0, 0);
    }
    __syncthreads();
    for (int kk = 0; kk < 64; kk += 4) {
      v2f a;  // consecutive dwords within padded row -> ds_load_2addr
      a.x = as[mtile * 16 + lm][kk + 2 * hh];
      a.y = as[mtile * 16 + lm][kk + 2 * hh + 1];
#pragma unroll
      for (int t = 0; t < 4; t++) {
        int j = (ntb + t) * 16 + lm;
        v2f bb;
        bb.x = Wt[kk + 2 * hh][j];
        bb.y = Wt[kk + 2 * hh + 1][j];
        acc[t] = wmma4(a, bb, acc[t]);
      }
    }
  }
#pragma unroll
  for (int t = 0; t < 4; t++) {
    int j = j0 + (ntb + t) * 16 + lm;
    float bv = bias[j];
#pragma unroll
    for (int r = 0; r < 8; r++) {
      float v = acc[t][r] + bv;
      if (thresh) v = (v > 0.5f) ? v : 0.0f;
      C[(size_t)(mtile * 16 + r + 8 * hh) * HSc + j] = v;
    }
  }
}

// QKV projections: grid (jc=32, wsel=3)
__global__ __launch_bounds__(256) void k_qkv(
    const float* __restrict__ A, const float* __restrict__ Wq,
    const float* __restrict__ Wk, const float* __restrict__ Wv,
    const float* __restrict__ bq, const float* __restrict__ bk,
    const float* __restrict__ bv, float* __restrict__ Cq,
    float* __restrict__ Ck, float* __restrict__ Cv) {
  __shared__ float as[64][65];
  __shared__ float Wt[64][JT];
  const int wsel = blockIdx.y;
  const float* W = (wsel == 0) ? Wq : (wsel == 1) ? Wk : Wv;
  const float* bia = (wsel == 0) ? bq : (wsel == 1) ? bk : bv;
  float* C = (wsel == 0) ? Cq : (wsel == 1) ? Ck : Cv;
  gemm64_body(A, W, bia, C, blockIdx.x * JT, false, as, Wt);
}

// Output projection + threshold: grid (jc=32)
__global__ __launch_bounds__(256) void k_out(const float* __restrict__ A,
                                             const float* __restrict__ W,
                                             const float* __restrict__ bias,
                                             float* __restrict__ C) {
  __shared__ float as[64][65];
  __shared__ float Wt[64][JT];
  gemm64_body(A, W, bias, C, blockIdx.x * JT, true, as, Wt);
}

// ------- attention core (L=4 tokens, dh=256): grid (h=16, b=16) -------
__global__ __launch_bounds__(256) void k_attn(const float* __restrict__ q,
                                              const float* __restrict__ k,
                                              const float* __restrict__ v,
                                              float* __restrict__ ctx) {
  const int h = blockIdx.x, b = blockIdx.y;
  __shared__ float qs[Lc][DHc], ks[Lc][DHc], vs[Lc][DHc];
  __shared__ float sc[Lc][Lc], at[Lc][Lc];
  const int tid = threadIdx.x;
  for (int i = tid; i < Lc * DHc; i += 256) {
    int t = i >> 8, d = i & 255;
    size_t idx = (size_t)(b * Lc + t) * HSc + h * DHc + d;
    qs[t][d] = q[idx];
    ks[t][d] = k[idx];
    vs[t][d] = v[idx];
  }
  if (tid < 16) sc[tid >> 2][tid & 3] = 0.0f;
  __syncthreads();
  {  // 16 (t,m) pairs x 16 partial segments of 16 dims
    int pair = tid >> 4, seg = tid & 15;
    int tt = pair >> 2, mm = pair & 3;
    float p = 0.0f;
    for (int d = seg * 16; d < seg * 16 + 16; d++) p += qs[tt][d] * ks[mm][d];
    atomicAdd(&sc[tt][mm], p);
  }
  __syncthreads();
  if (tid < Lc) {
    const float scale = 1.0f / 16.0f;  // 1/sqrt(256)
    float mx = -__builtin_inff();
    for (int mm = 0; mm < Lc; mm++) {
      float s = sc[tid][mm] * scale;
      if (s > mx) mx = s;
    }
    float e[Lc], sum = 0.0f;
    for (int mm = 0; mm < Lc; mm++) {
      e[mm] = __expf(sc[tid][mm] * scale - mx);
      sum += e[mm];
    }
    for (int mm = 0; mm < Lc; mm++) at[tid][mm] = e[mm] / sum;
  }
  __syncthreads();
  for (int i = tid; i < Lc * DHc; i += 256) {
    int t = i >> 8, d = i & 255;
    float o = 0.0f;
    for (int mm = 0; mm < Lc; mm++) o += at[t][mm] * vs[mm][d];
    ctx[(size_t)(b * Lc + t) * HSc + h * DHc + d] = o;
  }
}

// ---------------- launch ----------------
extern "C" void kernel_launch(void* const* d_in, const int* in_sizes, int n_in,
                              void* d_out, int out_size, void* d_ws,
                              size_t ws_size, hipStream_t stream) {
  (void)in_sizes; (void)n_in; (void)out_size; (void)ws_size;
  const float* x    = (const float*)d_in[0];
  const float* Wmlp = (const float*)d_in[1];
  const float* bmlp = (const float*)d_in[2];
  const float* Wq = (const float*)d_in[3];
  const float* bq = (const float*)d_in[4];
  const float* Wk = (const float*)d_in[5];
  const float* bk = (const float*)d_in[6];
  const float* Wv = (const float*)d_in[7];
  const float* bv = (const float*)d_in[8];
  const float* Wo = (const float*)d_in[9];
  const float* bo = (const float*)d_in[10];

  float* ws = (float*)d_ws;
  const size_t NTOK = (size_t)Bc * Lc * DMc * DMc;  // 262144
  float* G   = ws;                  // [B*L][64][64]
  float* S   = G + NTOK;            // [B*L][64]
  float* tok = S + (size_t)Bc * Lc * DMc;
  float* qb  = tok + NTOK;          // [64][4096] each
  float* kb  = qb + NTOK;
  float* vb  = kb + NTOK;
  float* ctx = vb + NTOK;

  const int nz = (int)(NTOK + Bc * Lc * DMc);
  k_zero<<<(nz + 255) / 256, 256, 0, stream>>>(G, nz);
  k_gram0<<<dim3(Pc / PT0, Bc), 256, 0, stream>>>(x, G, S);
  k_mlp_gram<<<dim3(HIDc / JT, 3, Bc), 256, 0, stream>>>(x, Wmlp, bmlp, G, S);
  k_adj<<<Bc * Lc, 64, 0, stream>>>(G, S, tok);
  k_qkv<<<dim3(HSc / JT, 3), 256, 0, stream>>>(tok, Wq, Wk, Wv, bq, bk, bv,
                                               qb, kb, vb);
  k_attn<<<dim3(NHc, Bc), 256, 0, stream>>>(qb, kb, vb, ctx);
  k_out<<<dim3(HSc / JT), 256, 0, stream>>>(ctx, Wo, bo, (float*)d_out);
}